// Adaptive_Node_Scale_GCN_14955076125148
// MI455X (gfx1250) — compile-verified
//
#include <hip/hip_runtime.h>
#include <hip/hip_bf16.h>

// ---------------------------------------------------------------------------
// Fused Adaptive_Node_Scale_GCN for gfx1250 (MI455X).
//   - fp32 WMMA (v_wmma_f32_16x16x4_f32): exact fp32 matrix math on matrix pipe
//   - fused diffusion (6 hops) + channel MLP: avoids 448 MiB "cat" intermediate
//   - adjacency AND hop2 scratch stored row-pair-INTERLEAVED in LDS ->
//     B-frags are single contiguous ds_load_b64 (mergeable to 2addr_b64)
//   - output routed through LDS so stores are 32B-contiguous along l
// ---------------------------------------------------------------------------

typedef float v2f __attribute__((ext_vector_type(2)));
typedef float v8f __attribute__((ext_vector_type(8)));

#define WMMA_F32X4(a, b, c) \
    __builtin_amdgcn_wmma_f32_16x16x4_f32(false, (a), false, (b), (short)0, (c), false, false)

// Problem constants (from reference config)
#define CB 16    // batch
#define CC 16    // channels
#define CD 4     // features
#define CK 128   // nodes
#define CL 128   // length
#define CE 10    // embedding

// LDS layout (floats)
// Adjacency: row-pair interleaved. (v,w) -> (v>>1)*ADJ2_STRIDE + 2*w + (v&1).
// ADJ2_STRIDE % 64 == 32 -> the two half-waves hit disjoint bank halves on
// B-frag b64 reads; staging writes (stride 2) hit 32 distinct banks.
#define ADJ2_STRIDE 288
#define ADJ_SIZE    (64 * ADJ2_STRIDE)        // 18432
#define WB_STRIDE   116            // gcd(116,64)=4 -> conflict-free A-frag b64 loads
#define WB_SIZE     (16 * WB_STRIDE)          // 1856
#define XB_STRIDE   132            // gcd(132,64)=4, even -> conflict-free b64 A-frags
#define SC2_STRIDE  40             // sc row-pair interleaved: 8 pairs x 40 = 320
#define PW_XO       0
#define PW_H1O      (16 * XB_STRIDE)          // 2112
#define PW_SCO      (2 * 16 * XB_STRIDE)      // 4224
#define PW_STRIDE   4552                       // 4544 used, %64==8 to stagger banks
#define PW_BASE     (ADJ_SIZE + WB_SIZE)       // 20288
#define LDS_FLOATS  (PW_BASE + 8 * PW_STRIDE)  // 56704 -> 226816 bytes

// ---------------------------------------------------------------------------
// Kernel 0: transpose shared supports: supT[s][v][w] = supports[s][w][v]
// ---------------------------------------------------------------------------
__global__ void k_transpose_supports(const float* __restrict__ sup,
                                     float* __restrict__ supT) {
    int s = blockIdx.x >> 7;
    int w = blockIdx.x & 127;
    int v = threadIdx.x;
    supT[s * CK * CK + v * CK + w] = sup[s * CK * CK + w * CK + v];
}

// ---------------------------------------------------------------------------
// Kernel 1: adaptive adjacency, stored transposed.
//   imp[f,w,e]   = sum_k imp_W[w,k] nv1[f,k,e] + imp_b[w]
//   score[f,w,j] = sum_e nv1[f,w,e]*imp[f,w,e]*nv2[f,e,j]
//   adp          = softmax_j(relu(score));  adpT[f][j][w] = adp[f][w][j]
// One block per (f,w); 128 threads (j).
// ---------------------------------------------------------------------------
__global__ void k_build_adp(const float* __restrict__ nv1,
                            const float* __restrict__ nv2,
                            const float* __restrict__ impW,
                            const float* __restrict__ impB,
                            float* __restrict__ adpT) {
    int f = blockIdx.x >> 7;
    int w = blockIdx.x & 127;
    int j = threadIdx.x;

    __shared__ float impe[CE];
    __shared__ float red[CK];

    if (j < CE) {
        float acc = impB[w];
        for (int k = 0; k < CK; ++k)
            acc += impW[w * CK + k] * nv1[(f * CK + k) * CE + j];
        impe[j] = acc;
    }
    __syncthreads();

    float sscore = 0.f;
    #pragma unroll
    for (int e = 0; e < CE; ++e)
        sscore += nv1[(f * CK + w) * CE + e] * impe[e] * nv2[(f * CE + e) * CK + j];
    sscore = fmaxf(sscore, 0.f);

    // block softmax over j
    red[j] = sscore;
    __syncthreads();
    for (int off = 64; off > 0; off >>= 1) {
        if (j < off) red[j] = fmaxf(red[j], red[j + off]);
        __syncthreads();
    }
    float mx = red[0];
    __syncthreads();
    float ex = expf(sscore - mx);
    red[j] = ex;
    __syncthreads();
    for (int off = 64; off > 0; off >>= 1) {
        if (j < off) red[j] += red[j + off];
        __syncthreads();
    }
    float denom = red[0];

    adpT[f * CK * CK + j * CK + w] = ex / denom;
}

// ---------------------------------------------------------------------------
// MLP accumulate: Y[o][k-tiles] += W[:, col0:col0+16] @ buf   (buf is 16xK,
// row-major with stride st in LDS). ks-outer so one A-frag feeds 8 WMMAs.
// ---------------------------------------------------------------------------
__device__ __forceinline__ void mlp_accum(const float* wb, const float* buf,
                                          int st, int col0, int m, int h,
                                          v8f (&Y)[8]) {
    #pragma unroll
    for (int ks = 0; ks < 4; ++ks) {
        int cc = ks * 4 + h * 2;
        v2f a = *(const v2f*)&wb[m * WB_STRIDE + col0 + cc];
        #pragma unroll
        for (int ct = 0; ct < 8; ++ct) {
            v2f bb;
            bb.x = buf[cc * st + ct * 16 + m];
            bb.y = buf[(cc + 1) * st + ct * 16 + m];
            Y[ct] = WMMA_F32X4(a, bb, Y[ct]);
        }
    }
}

// ---------------------------------------------------------------------------
// Main fused kernel. Grid = D*B*(L/8) = 1024 blocks, 256 threads (8 waves).
// Block = (f, b, l-chunk of 8); wave ww handles l = l0 + ww.
// ---------------------------------------------------------------------------
__global__ void __launch_bounds__(256)
k_gcn_main(const float* __restrict__ x, const float* __restrict__ mlpW,
           const float* __restrict__ mlpB, const float* __restrict__ supT,
           const float* __restrict__ adpT, float* __restrict__ out) {
    extern __shared__ float lds[];

    const int tid  = threadIdx.x;
    const int lane = tid & 31;
    const int wv   = tid >> 5;      // wave index 0..7
    const int m    = lane & 15;     // row/col within 16x16 tile
    const int h    = lane >> 4;     // half-wave 0/1

    const int blk = blockIdx.x;
    const int lc  = blk & 15;           // l-chunk
    const int b   = (blk >> 4) & 15;    // batch
    const int f   = blk >> 8;           // feature
    const int l0  = lc * 8;

    float* adj = lds;                       // interleaved [64 pairs][288]
    float* wb  = lds + ADJ_SIZE;            // [16][116]
    float* myw = lds + PW_BASE + wv * PW_STRIDE;
    float* xb  = myw + PW_XO;               // X    [16 c][132] row-major
    float* h1  = myw + PW_H1O;              // hop1 [16 c][132] row-major
    float* sc  = myw + PW_SCO;              // x2 coltile scratch, interleaved [8][40]

    // ---- stage W_f (16 x 112) ----
    for (int i = tid; i < 16 * 112; i += 256)
        wb[(i / 112) * WB_STRIDE + (i % 112)] = mlpW[f * 16 * 112 + i];

    // ---- cooperative X stage for all 8 l's (coalesced 32B global reads) ----
    {
        int li  = tid & 7;                       // which wave's buffer / l
        int cv0 = tid >> 3;                      // 0..31
        float* dst = lds + PW_BASE + li * PW_STRIDE + PW_XO;
        for (int it = 0; it < 64; ++it) {
            int idx = cv0 + it * 32;             // 0..2047 -> (c, v)
            int c = idx >> 7, v = idx & 127;
            dst[c * XB_STRIDE + v] =
                x[(((b * 16 + c) * 4 + f) << 14) + v * 128 + l0 + li];
        }
    }

    // ---- init Y accumulators with bias (broadcast over k) ----
    const v8f vzero = {};
    v8f Y[8];
    float bias[8];
    #pragma unroll
    for (int r = 0; r < 8; ++r) bias[r] = mlpB[f * 16 + r + 8 * h];
    #pragma unroll
    for (int ct = 0; ct < 8; ++ct) {
        v8f t = vzero;
        #pragma unroll
        for (int r = 0; r < 8; ++r) t[r] = bias[r];
        Y[ct] = t;
    }

    // ---- supports: s = 0,1 shared; s = 2 adaptive ----
    for (int s = 0; s < 3; ++s) {
        __syncthreads();  // previous compute done before overwriting adj
        const float* asrc = (s < 2) ? (supT + s * CK * CK) : (adpT + f * CK * CK);
        // interleave row pairs: (v,w) -> pair v>>1, 2*w + (v&1)
        for (int i = tid; i < CK * CK; i += 256) {
            int v = i >> 7, w = i & 127;
            adj[(v >> 1) * ADJ2_STRIDE + w * 2 + (v & 1)] = asrc[i];
        }
        __syncthreads();  // adj + X staged -> compute

        // block 0 of cat is X itself (count it once)
        if (s == 0) mlp_accum(wb, xb, XB_STRIDE, 0, m, h, Y);

        // ---- hop1: h1 = X @ A^T ; ks-outer, 8 independent acc chains ----
        {
            v8f hacc[8];
            #pragma unroll
            for (int ct = 0; ct < 8; ++ct) hacc[ct] = vzero;
            #pragma unroll 2
            for (int ks = 0; ks < 32; ++ks) {
                int kk   = ks * 4 + h * 2;          // A contraction offset
                int pair = ks * 2 + h;              // adjacency row pair
                v2f a = *(const v2f*)&xb[m * XB_STRIDE + kk];
                #pragma unroll
                for (int ct = 0; ct < 8; ++ct) {
                    v2f bb = *(const v2f*)&adj[pair * ADJ2_STRIDE + (ct * 16 + m) * 2];
                    hacc[ct] = WMMA_F32X4(a, bb, hacc[ct]);
                }
            }
            #pragma unroll
            for (int ct = 0; ct < 8; ++ct)
                #pragma unroll
                for (int r = 0; r < 8; ++r)
                    h1[(r + 8 * h) * XB_STRIDE + ct * 16 + m] = hacc[ct][r];
        }

        // MLP block (1 + 2s) from h1 (same-wave DS ops are in order)
        mlp_accum(wb, h1, XB_STRIDE, (1 + 2 * s) * 16, m, h, Y);

        // ---- hop2 tile-by-tile, fold straight into MLP block (2 + 2s) ----
        {
            int col0 = (2 + 2 * s) * 16;
            #pragma unroll
            for (int ct = 0; ct < 8; ++ct) {
                v8f acc = vzero;
                #pragma unroll 2
                for (int ks = 0; ks < 32; ++ks) {
                    int kk   = ks * 4 + h * 2;
                    int pair = ks * 2 + h;
                    v2f a = *(const v2f*)&h1[m * XB_STRIDE + kk];
                    v2f bb = *(const v2f*)&adj[pair * ADJ2_STRIDE + (ct * 16 + m) * 2];
                    acc = WMMA_F32X4(a, bb, acc);
                }
                // cross-lane relayout D-frag -> B-frag via per-wave scratch,
                // stored row-pair interleaved so reads are single b64 loads
                #pragma unroll
                for (int r = 0; r < 8; ++r)
                    sc[((r >> 1) + 4 * h) * SC2_STRIDE + m * 2 + (r & 1)] = acc[r];
                #pragma unroll
                for (int ks = 0; ks < 4; ++ks) {
                    int cpair = ks * 2 + h;         // channel row pair (cc>>1)
                    v2f a = *(const v2f*)&wb[m * WB_STRIDE + col0 + ks * 4 + h * 2];
                    v2f bb = *(const v2f*)&sc[cpair * SC2_STRIDE + m * 2];
                    Y[ct] = WMMA_F32X4(a, bb, Y[ct]);
                }
            }
        }
    }

    // ---- park Y into h1 (free now), then cooperative coalesced writeout ----
    #pragma unroll
    for (int ct = 0; ct < 8; ++ct)
        #pragma unroll
        for (int r = 0; r < 8; ++r)
            h1[(r + 8 * h) * XB_STRIDE + ct * 16 + m] = Y[ct][r];
    __syncthreads();

    {
        int li  = tid & 7;    // l within chunk: 8 consecutive threads -> 32B burst
        int grp = tid >> 3;   // 0..31 over (o,k) pairs
        const float* src = lds + PW_BASE + li * PW_STRIDE + PW_H1O;
        for (int it = 0; it < 64; ++it) {
            int p = grp + it * 32;         // 0..2047
            int o = p >> 7, k = p & 127;
            out[(((b * 16 + o) * 4 + f) << 14) + k * 128 + l0 + li] =
                src[o * XB_STRIDE + k];
        }
    }
}

// ---------------------------------------------------------------------------
extern "C" void kernel_launch(void* const* d_in, const int* in_sizes, int n_in,
                              void* d_out, int out_size, void* d_ws, size_t ws_size,
                              hipStream_t stream) {
    (void)in_sizes; (void)n_in; (void)out_size; (void)ws_size;

    const float* x    = (const float*)d_in[0];
    const float* sup  = (const float*)d_in[1];
    const float* nv1  = (const float*)d_in[2];
    const float* nv2  = (const float*)d_in[3];
    const float* impW = (const float*)d_in[4];
    const float* impB = (const float*)d_in[5];
    const float* mlpW = (const float*)d_in[6];
    const float* mlpB = (const float*)d_in[7];
    float* out = (float*)d_out;

    float* supT = (float*)d_ws;               // 2 * 128 * 128
    float* adpT = supT + 2 * CK * CK;         // 4 * 128 * 128

    k_transpose_supports<<<2 * CK, CK, 0, stream>>>(sup, supT);
    k_build_adp<<<CD * CK, CK, 0, stream>>>(nv1, nv2, impW, impB, adpT);

    size_t shmem = (size_t)LDS_FLOATS * sizeof(float);  // 226,816 B (<= 320KB/WGP)
    k_gcn_main<<<CD * CB * (CL / 8), 256, shmem, stream>>>(x, mlpW, mlpB,
                                                           supT, adpT, out);
}